// WENO_temporal_21534966022518
// MI455X (gfx1250) — compile-verified
//
#include <hip/hip_runtime.h>
#include <math.h>

// ---------------------------------------------------------------------------
// Learned-WENO SSP-RK3 step for MI455X (gfx1250).
//  - f32 stencil: bandwidth/VALU bound (~100MB traffic => ~5us at 23.3TB/s).
//  - conv1 (K=10) done with V_WMMA_F32_16X16X4_F32 (exact f32 matmul).
//  - halo tiles moved Global->LDS with global_load_async_to_lds_b32 (ASYNCcnt).
// ---------------------------------------------------------------------------

#define TILE   256
#define WENO_N 131072
#define EPSF   1e-40f
#define DTF    2e-05f

typedef __attribute__((ext_vector_type(2))) float v2f;
typedef __attribute__((ext_vector_type(8))) float v8f;

#if defined(__HIP_DEVICE_COMPILE__) && __has_builtin(__builtin_amdgcn_wmma_f32_16x16x4_f32)
#define HAVE_WMMA4 1
#else
#define HAVE_WMMA4 0
#endif

// jax.nn.elu uses expm1 (not exp(x)-1): match it.
__device__ __forceinline__ float elu1(float x) {
  return x > 0.0f ? x : expm1f(x);
}

// CDNA5 async global->LDS copy (one f32 per lane), tracked by ASYNCcnt.
__device__ __forceinline__ void async_ld_f32(float* lds_dst, const float* gsrc) {
  unsigned lofs = (unsigned)(unsigned long long)(uintptr_t)lds_dst; // LDS byte addr = addr[31:0]
  asm volatile("global_load_async_to_lds_b32 %0, %1, off"
               :: "v"(lofs), "v"(gsrc) : "memory");
}

__device__ __forceinline__ void wait_async0() {
#if defined(__HIP_DEVICE_COMPILE__) && __has_builtin(__builtin_amdgcn_s_wait_asynccnt)
  __builtin_amdgcn_s_wait_asynccnt(0);
#else
  asm volatile("s_wait_asynccnt 0x0" ::: "memory");
#endif
}

// WENO-JS reconstruction with learned beta scaling (matches _recon_minus).
__device__ __forceinline__ float recon5(const float u0[5], float w0, float w1, float w2) {
  const float c1312 = 13.0f / 12.0f;
  float s10 = u0[0] - 2.0f * u0[1] + u0[2];
  float s11 = u0[1] - 2.0f * u0[2] + u0[3];
  float s12 = u0[2] - 2.0f * u0[3] + u0[4];
  float t20 = u0[0] - 4.0f * u0[1] + 3.0f * u0[2];
  float t21 = u0[1] - u0[3];
  float t22 = 3.0f * u0[2] - 4.0f * u0[3] + u0[4];
  float b0 = (c1312 * s10 * s10 + 0.25f * t20 * t20) * (w0 + 0.1f);
  float b1 = (c1312 * s11 * s11 + 0.25f * t21 * t21) * (w1 + 0.1f);
  float b2 = (c1312 * s12 * s12 + 0.25f * t22 * t22) * (w2 + 0.1f);
  float r0 = 1.0f / (b0 + EPSF);
  float r1 = 1.0f / (b1 + EPSF);
  float r2 = 1.0f / (b2 + EPSF);
  float a0 = r0 * r0 * 0.1f;
  float a1 = r1 * r1 * 0.6f;
  float a2 = r2 * r2 * 0.3f;
  float inv = 1.0f / (a0 + a1 + a2);
  float f0 = ( 2.0f * u0[0] - 7.0f * u0[1] + 11.0f * u0[2]) * (1.0f / 6.0f);
  float f1 = (-u0[1] + 5.0f * u0[2] + 2.0f * u0[3])          * (1.0f / 6.0f);
  float f2 = ( 2.0f * u0[2] + 5.0f * u0[3] - u0[4])          * (1.0f / 6.0f);
  return (a0 * f0 + a1 * f1 + a2 * f2) * inv;
}

// mode 0: out = v - DT*d          (v = u)
// mode 1: out = (3u + v - DT*d)/4 (v = u1)
// mode 2: out = (u + 2v - 2DT*d)/3 (v = u2)
__global__ __launch_bounds__(256) void weno_stage_kernel(
    const float* __restrict__ ub, const float* __restrict__ v,
    float* __restrict__ out,
    const float* __restrict__ W1, const float* __restrict__ b1,
    const float* __restrict__ W2, const float* __restrict__ b2,
    const float* __restrict__ W3, const float* __restrict__ b3,
    int Nn, int mode)
{
  __shared__ float sUe1[TILE + 16];        // Ue tile (local), e in [T0-4, T0+TILE+10)
  __shared__ float sUe2[TILE + 16];        // Ue tile (mirrored)
  __shared__ float sY[(TILE + 24) * 2];    // d1,d2 (zero-padded tail for WMMA groups)
  __shared__ float sH1[(TILE + 16) * 5];   // conv1 output
  __shared__ float sBW1[TILE + 4];         // bw for fhm (local)
  __shared__ float sBW2[TILE + 4];         // bw for fhp (mirrored)
  __shared__ float sF[TILE + 4];           // interface fluxes Fhat
  __shared__ float sWt[140];               // packed weights

  const int tid  = threadIdx.x;
  const int T0   = blockIdx.x * TILE;
  const size_t boff = (size_t)blockIdx.y * (size_t)Nn;
  const float* vb = v + boff;
  const int Q2 = Nn - T0 - TILE + 1;       // base of mirrored bw range

  // ---- weights -> LDS: [0,50)=W1, [50,55)=b1, [55,130)=W2, [130,133)=b2,
  //                      [133,136)=W3, [136]=b3
  for (int l = tid; l < 137; l += TILE) {
    float w;
    if      (l < 50)  w = W1[l];
    else if (l < 55)  w = b1[l - 50];
    else if (l < 130) w = W2[l - 55];
    else if (l < 133) w = b2[l - 130];
    else if (l < 136) w = W3[l - 133];
    else              w = b3[0];
    sWt[l] = w;
  }

  // ---- async halo loads for both Ue tiles (Ue[e] = v[(e-3) mod N])
  for (int pass = 0; pass < 2; ++pass) {
    int Q = pass ? Q2 : T0;
    float* ue = pass ? sUe2 : sUe1;
    for (int l = tid; l < TILE + 14; l += TILE) {
      int e = Q - 4 + l;
      if (e >= 0 && e < Nn + 6) {
        int g = e - 3;
        if (g < 0) g += Nn; else if (g >= Nn) g -= Nn;
        async_ld_f32(&ue[l], &vb[g]);
      } else {
        ue[l] = 0.0f;   // never read (pad semantics live at y/h1 level)
      }
    }
  }
  wait_async0();
  __syncthreads();

  const int lane = tid & 31;
  const int wave = tid >> 5;
  const int hi   = lane >> 4;   // lane half (K-half selector for A/B frags)
  const int c16  = lane & 15;   // M (for A) / N channel (for B,D)

#if HAVE_WMMA4
  // B fragments: W1 as 4x16 f32 B-matrix per K-chunk. VGPR g: lanes0-15 K=g,
  // lanes16-31 K=g+2 (mirrors documented 16x4 f32 A layout). K>=10 / c>=5
  // columns are zeroed HERE, so A fragments never need masking (x*0 == 0,
  // all y values finite) -> clean ds_load_b64 + wmma sequences.
  v2f bfr[3];
  float b1c = (c16 < 5) ? sWt[50 + c16] : 0.0f;
  #pragma unroll
  for (int kk = 0; kk < 3; ++kk) {
    int k0 = kk * 4 + 2 * hi;              // K index of .x ; .y = k0+1
    bool valid = (k0 < 10) && (c16 < 5);   // K=10 real taps, 5 real channels
    bfr[kk].x = valid ? sWt[k0 * 5 + c16]       : 0.0f;
    bfr[kk].y = valid ? sWt[(k0 + 1) * 5 + c16] : 0.0f;
  }
#endif

  // =======================  network, two tiles  ==========================
  for (int pass = 0; pass < 2; ++pass) {
    int Q = pass ? Q2 : T0;
    const float* ue = pass ? sUe2 : sUe1;
    float* bw = pass ? sBW2 : sBW1;

    // ---- y = (d1, d2); zero outside [0, N+4) (reference zero-pads convs)
    for (int l = tid; l < TILE + 24; l += TILE) {
      float d1 = 0.0f, d2 = 0.0f;
      if (l < TILE + 12) {
        int mp = Q - 4 + l;
        if (mp >= 0 && mp < Nn + 4) {
          float a = ue[l], c = ue[l + 2];
          d1 = c - a;
          d2 = c - 2.0f * ue[l + 1] + a;
        }
      }
      sY[l * 2 + 0] = d1;
      sY[l * 2 + 1] = d2;
    }
    __syncthreads();

    // ---- conv1: h1[m][c] = elu(b1 + sum_{k<10} yw[m][k] * W1[k][c])
#if HAVE_WMMA4
    for (int g = wave; g < 17; g += 8) {           // wave-uniform: EXEC all-1s
      int lbase = g * 16 + c16;                    // position (M) for this lane
      v8f acc = {b1c, b1c, b1c, b1c, b1c, b1c, b1c, b1c};
      #pragma unroll
      for (int kk = 0; kk < 3; ++kk) {
        int t = kk * 2 + hi;                       // tap index; pairs (in0,in1) adjacent
        v2f a;                                     // unmasked: B is 0 for K>=10
        a.x = sY[(lbase + t) * 2 + 0];
        a.y = sY[(lbase + t) * 2 + 1];
        acc = __builtin_amdgcn_wmma_f32_16x16x4_f32(
            false, a, false, bfr[kk], (short)0, acc, false, false);
      }
      if (c16 < 5) {
        #pragma unroll
        for (int r = 0; r < 8; ++r) {
          int lh = g * 16 + r + 8 * hi;            // D: VGPR r -> M=r (+8 upper half)
          int m  = Q - 2 + lh;
          float val = (m >= 0 && m < Nn + 4) ? elu1(acc[r]) : 0.0f;
          sH1[lh * 5 + c16] = val;
        }
      }
    }
#else
    for (int l = tid; l < TILE + 8; l += TILE) {
      int m = Q - 2 + l;
      if (m >= 0 && m < Nn + 4) {
        float acc[5];
        #pragma unroll
        for (int c = 0; c < 5; ++c) acc[c] = sWt[50 + c];
        #pragma unroll
        for (int t = 0; t < 5; ++t) {
          float y0 = sY[(l + t) * 2 + 0];
          float y1 = sY[(l + t) * 2 + 1];
          #pragma unroll
          for (int c = 0; c < 5; ++c)
            acc[c] += y0 * sWt[(t * 2 + 0) * 5 + c] + y1 * sWt[(t * 2 + 1) * 5 + c];
        }
        #pragma unroll
        for (int c = 0; c < 5; ++c) sH1[l * 5 + c] = elu1(acc[c]);
      } else {
        #pragma unroll
        for (int c = 0; c < 5; ++c) sH1[l * 5 + c] = 0.0f;
      }
    }
#endif
    __syncthreads();

    // ---- conv2 (5x5x3) + elu + conv3 (1x3x1) + sigmoid -> bw
    for (int l = tid; l < TILE + 4; l += TILE) {
      float h20 = sWt[130], h21 = sWt[131], h22 = sWt[132];
      #pragma unroll
      for (int t = 0; t < 5; ++t) {
        #pragma unroll
        for (int c = 0; c < 5; ++c) {
          float hv = sH1[(l + t) * 5 + c];
          int wbase = 55 + (t * 5 + c) * 3;
          h20 += hv * sWt[wbase + 0];
          h21 += hv * sWt[wbase + 1];
          h22 += hv * sWt[wbase + 2];
        }
      }
      float z = sWt[136] + elu1(h20) * sWt[133] + elu1(h21) * sWt[134]
                         + elu1(h22) * sWt[135];
      bw[l] = 1.0f / (1.0f + expf(-z));
    }
    __syncthreads();
  }

  // =======================  flux at interfaces  ==========================
  for (int l = tid; l < TILE + 1; l += TILE) {
    // interface j = T0 + l; Ue window Ue[j..j+5] = sUe1[l+4 .. l+9]
    float uf0 = sUe1[l + 4], uf1 = sUe1[l + 5], uf2 = sUe1[l + 6];
    float uf3 = sUe1[l + 7], uf4 = sUe1[l + 8], uf5 = sUe1[l + 9];
    float M = fmaxf(fabsf(uf0), fabsf(uf1));
    M = fmaxf(M, fabsf(uf2)); M = fmaxf(M, fabsf(uf3));
    M = fmaxf(M, fabsf(uf4)); M = fmaxf(M, fabsf(uf5));
    float q0 = 0.5f * uf0 * uf0, q1 = 0.5f * uf1 * uf1, q2 = 0.5f * uf2 * uf2;
    float q3 = 0.5f * uf3 * uf3, q4 = 0.5f * uf4 * uf4, q5 = 0.5f * uf5 * uf5;
    float u0m[5] = { 0.5f * (q0 + M * uf0), 0.5f * (q1 + M * uf1),
                     0.5f * (q2 + M * uf2), 0.5f * (q3 + M * uf3),
                     0.5f * (q4 + M * uf4) };
    float fhm = recon5(u0m, sBW1[l], sBW1[l + 1], sBW1[l + 2]);
    // fm[5..1] reversed; mirrored weights w_k = bw[N+3-j-k] -> sBW2[TILE+2-l-k]
    float u0p[5] = { 0.5f * (q5 - M * uf5), 0.5f * (q4 - M * uf4),
                     0.5f * (q3 - M * uf3), 0.5f * (q2 - M * uf2),
                     0.5f * (q1 - M * uf1) };
    int mb = TILE + 2 - l;
    float fhp = recon5(u0p, sBW2[mb], sBW2[mb - 1], sBW2[mb - 2]);
    sF[l] = fhm + fhp;
  }
  __syncthreads();

  // =======================  deriv + RK combine  ==========================
  {
    const float DXF = (float)(2.0 * 3.14159265358979323846 / 131072.0);
    int i = T0 + tid;
    float d  = (sF[tid + 1] - sF[tid]) / DXF;
    float vv = sUe1[tid + 7];                 // v[i] = Ue[i+3]
    float ubv = (mode != 0) ? ub[boff + i] : 0.0f;
    float r;
    if      (mode == 0) r = vv - DTF * d;
    else if (mode == 1) r = (3.0f * ubv + vv - DTF * d) * 0.25f;
    else                r = (ubv + 2.0f * vv - 2.0f * DTF * d) / 3.0f;
    out[boff + i] = r;
  }
}

extern "C" void kernel_launch(void* const* d_in, const int* in_sizes, int n_in,
                              void* d_out, int out_size, void* d_ws, size_t ws_size,
                              hipStream_t stream) {
  (void)n_in; (void)out_size; (void)ws_size;
  const float* u  = (const float*)d_in[0];
  const float* W1 = (const float*)d_in[1];
  const float* b1 = (const float*)d_in[2];
  const float* W2 = (const float*)d_in[3];
  const float* b2 = (const float*)d_in[4];
  const float* W3 = (const float*)d_in[5];
  const float* b3 = (const float*)d_in[6];
  float* out = (float*)d_out;
  float* u2  = (float*)d_ws;          // 16MB scratch for stage-2 field

  const int N = WENO_N;
  const int B = in_sizes[0] / N;
  dim3 grid(N / TILE, B), blk(TILE);

  // u1 lives in d_out during stages 1-2, final result overwrites it in stage 3.
  weno_stage_kernel<<<grid, blk, 0, stream>>>(u, u,   out, W1, b1, W2, b2, W3, b3, N, 0);
  weno_stage_kernel<<<grid, blk, 0, stream>>>(u, out, u2,  W1, b1, W2, b2, W3, b3, N, 1);
  weno_stage_kernel<<<grid, blk, 0, stream>>>(u, u2,  out, W1, b1, W2, b2, W3, b3, N, 2);
}